// TotalCorrelationLoss_41403484733530
// MI455X (gfx1250) — compile-verified
//
#include <hip/hip_runtime.h>
#include <math.h>

typedef __attribute__((ext_vector_type(2))) float v2f;
typedef __attribute__((ext_vector_type(4))) float v4f;
typedef __attribute__((ext_vector_type(8))) float v8f;

#define NROWS 2048
#define KDIM  128
#define EPSV  1e-6f
#define ITILE 4   // rows per thread in the marginal kernel

// ---------------------------------------------------------------------------
// K1: per-column standardization (torch-style unbiased std, ddof=1).
// One block per column k; writes row-major xs and column-major xsT.
// ---------------------------------------------------------------------------
__global__ void __launch_bounds__(256)
k_standardize(const float* __restrict__ x, float* __restrict__ xs,
              float* __restrict__ xsT) {
  __shared__ float ssum[256];
  __shared__ float ssq[256];
  const int k = blockIdx.x;
  const int t = threadIdx.x;
  float s = 0.f, q = 0.f;
  for (int i = t; i < NROWS; i += 256) {
    float v = x[i * KDIM + k];
    s += v;
    q += v * v;
  }
  ssum[t] = s; ssq[t] = q;
  __syncthreads();
  for (int off = 128; off > 0; off >>= 1) {
    if (t < off) { ssum[t] += ssum[t + off]; ssq[t] += ssq[t + off]; }
    __syncthreads();
  }
  const float sum  = ssum[0];
  const float mean = sum / (float)NROWS;
  const float var  = (ssq[0] - sum * sum / (float)NROWS) / (float)(NROWS - 1);
  const float inv  = 1.0f / (sqrtf(var) + EPSV);
  for (int i = t; i < NROWS; i += 256) {
    float v = (x[i * KDIM + k] - mean) * inv;
    xs[i * KDIM + k]  = v;
    xsT[k * NROWS + i] = v;
  }
}

// ---------------------------------------------------------------------------
// K2: half squared row norms n_i = 0.5*||xs_i||^2; also zero S_joint for the
// atomic accumulation in k_joint.
// ---------------------------------------------------------------------------
__global__ void __launch_bounds__(256)
k_rownorm(const float* __restrict__ xs, float* __restrict__ nhalf,
          float* __restrict__ Sjoint) {
  const int i = blockIdx.x * blockDim.x + threadIdx.x;
  if (i >= NROWS) return;
  const v4f* row = (const v4f*)(xs + i * KDIM);
  float s = 0.f;
#pragma unroll
  for (int c = 0; c < KDIM / 4; ++c) {
    v4f v = row[c];
    s += v.x * v.x + v.y * v.y + v.z * v.z + v.w * v.w;
  }
  nhalf[i] = 0.5f * s;
  Sjoint[i] = 0.f;
}

// ---------------------------------------------------------------------------
// K3: joint term. G = X @ X^T via V_WMMA_F32_16X16X4_F32; fused
// exp(G_ij - n_i - n_j) and row-sum into S_joint.
// Block = (32,4): 4 waves, each wave owns one 16x16 output tile.
// A-matrix fp32 16x4 per-lane layout (ISA 7.12.2): lanes 0-15 hold M=lane,
// K={0,1}; lanes 16-31 hold M=lane-16, K={2,3}. B (4x16 of X^T) ends up as the
// same contiguous 2-float fetch from row j0+n of X.
// ---------------------------------------------------------------------------
__global__ void __launch_bounds__(128)
k_joint(const float* __restrict__ xs, const float* __restrict__ nhalf,
        float* __restrict__ Sjoint) {
  const int lane = threadIdx.x;                 // 0..31 (wave32)
  const int half = lane >> 4;                   // 0: K={0,1}, 1: K={2,3}
  const int m    = lane & 15;
  const int j0   = blockIdx.x * 16;
  const int i0   = (blockIdx.y * 4 + threadIdx.y) * 16;

  const float* arow = xs + (i0 + m) * KDIM + half * 2;
  const float* brow = xs + (j0 + m) * KDIM + half * 2;

  v8f c = {};
#pragma unroll 8
  for (int kk = 0; kk < KDIM; kk += 4) {
    v2f a = *(const v2f*)(arow + kk);
    v2f b = *(const v2f*)(brow + kk);
    c = __builtin_amdgcn_wmma_f32_16x16x4_f32(
        /*neg_a=*/false, a, /*neg_b=*/false, b,
        /*c_mod=*/(short)0, c, /*reuse_a=*/false, /*reuse_b=*/false);
  }

  const float ncol = nhalf[j0 + m];
  float rowsum[8];
#pragma unroll
  for (int r = 0; r < 8; ++r) {
    // C/D layout: vgpr r, lanes 0-15 -> M=r, lanes 16-31 -> M=r+8; N = lane&15
    const float nrow = nhalf[i0 + r + half * 8];
    float v = __expf(c[r] - nrow - ncol);
    // reduce across the 16 lanes of this half-group (columns of the tile)
    v += __shfl_xor(v, 1, 16);
    v += __shfl_xor(v, 2, 16);
    v += __shfl_xor(v, 4, 16);
    v += __shfl_xor(v, 8, 16);
    rowsum[r] = v;
  }
  if (m == 0) {
#pragma unroll
    for (int r = 0; r < 8; ++r)
      atomicAdd(&Sjoint[i0 + r + half * 8], rowsum[r]);
  }
}

// ---------------------------------------------------------------------------
// K4: marginal term. For column k: S_marg[i,k] = sum_j exp(-(x_ik-x_jk)^2/2).
// Column staged in LDS (8KB); per-iteration LDS read is a broadcast.
// Dominant cost: N*N*K = 537M v_exp_f32. Register-block ITILE=4 rows per
// thread so each LDS broadcast feeds 4 independent exp chains (TRANS32 pipe
// co-executes with the VALU sub/mul/add work; 4x less DS traffic).
// ---------------------------------------------------------------------------
__global__ void __launch_bounds__(256)
k_marginal(const float* __restrict__ xsT, float* __restrict__ logmarg) {
  __shared__ float col[NROWS];
  const int k = blockIdx.x;
  const int t = threadIdx.x;
  for (int j = t; j < NROWS; j += 256) col[j] = xsT[k * NROWS + j];
  __syncthreads();

  const int ibase = blockIdx.y * (256 * ITILE) + t;  // rows ibase + {0,256,512,768}
  const float xi0 = col[ibase +   0];
  const float xi1 = col[ibase + 256];
  const float xi2 = col[ibase + 512];
  const float xi3 = col[ibase + 768];

  float s0 = 0.f, s1 = 0.f, s2 = 0.f, s3 = 0.f;
#pragma unroll 4
  for (int j = 0; j < NROWS; ++j) {
    const float xj = col[j];
    float d0 = xi0 - xj; s0 += __expf(-0.5f * d0 * d0);
    float d1 = xi1 - xj; s1 += __expf(-0.5f * d1 * d1);
    float d2 = xi2 - xj; s2 += __expf(-0.5f * d2 * d2);
    float d3 = xi3 - xj; s3 += __expf(-0.5f * d3 * d3);
  }

  logmarg[(ibase +   0) * KDIM + k] = __logf(s0);
  logmarg[(ibase + 256) * KDIM + k] = __logf(s1);
  logmarg[(ibase + 512) * KDIM + k] = __logf(s2);
  logmarg[(ibase + 768) * KDIM + k] = __logf(s3);
}

// ---------------------------------------------------------------------------
// K5: tc = (K-1)*logN + mean_i( log S_joint[i] - sum_k logmarg[i,k] )
// (the Gaussian constants cancel exactly: c_joint == K*c_marg, sigma=1)
// ---------------------------------------------------------------------------
__global__ void __launch_bounds__(256)
k_finalize(const float* __restrict__ Sjoint, const float* __restrict__ logmarg,
           float* __restrict__ out) {
  __shared__ float red[256];
  const int t = threadIdx.x;
  float acc = 0.f;
  for (int i = t; i < NROWS; i += 256) {
    float term = __logf(Sjoint[i]);
    const v4f* lm = (const v4f*)(logmarg + i * KDIM);
    float sm = 0.f;
#pragma unroll
    for (int c = 0; c < KDIM / 4; ++c) {
      v4f v = lm[c];
      sm += v.x + v.y + v.z + v.w;
    }
    acc += term - sm;
  }
  red[t] = acc;
  __syncthreads();
  for (int off = 128; off > 0; off >>= 1) {
    if (t < off) red[t] += red[t + off];
    __syncthreads();
  }
  if (t == 0) {
    const float logN = __logf((float)NROWS);
    out[0] = (float)(KDIM - 1) * logN + red[0] / (float)NROWS;  // beta = 1
  }
}

// ---------------------------------------------------------------------------
extern "C" void kernel_launch(void* const* d_in, const int* in_sizes, int n_in,
                              void* d_out, int out_size, void* d_ws, size_t ws_size,
                              hipStream_t stream) {
  (void)in_sizes; (void)n_in; (void)out_size; (void)ws_size;
  const float* x = (const float*)d_in[0];

  float* ws      = (float*)d_ws;
  float* xs      = ws;                     // N*K   standardized, row-major
  float* xsT     = xs + NROWS * KDIM;      // N*K   standardized, col-major
  float* nhalf   = xsT + NROWS * KDIM;     // N     0.5*||x_i||^2
  float* Sjoint  = nhalf + NROWS;          // N     joint Gaussian row sums
  float* logmarg = Sjoint + NROWS;         // N*K   log marginal sums

  k_standardize<<<dim3(KDIM), dim3(256), 0, stream>>>(x, xs, xsT);
  k_rownorm<<<dim3(NROWS / 256), dim3(256), 0, stream>>>(xs, nhalf, Sjoint);
  k_joint<<<dim3(NROWS / 16, NROWS / 64), dim3(32, 4), 0, stream>>>(xs, nhalf, Sjoint);
  k_marginal<<<dim3(KDIM, NROWS / (256 * ITILE)), dim3(256), 0, stream>>>(xsT, logmarg);
  k_finalize<<<dim3(1), dim3(256), 0, stream>>>(Sjoint, logmarg, (float*)d_out);
}